// Fuse_Stage_37391985279460
// MI455X (gfx1250) — compile-verified
//
#include <hip/hip_runtime.h>
#include <hip/hip_bf16.h>

typedef __attribute__((ext_vector_type(2))) float v2f;
typedef __attribute__((ext_vector_type(4))) float v4f;
typedef __attribute__((ext_vector_type(8))) float v8f;

// Explicit global (addrspace 1) pointers: keeps every hot-loop access on the
// GLOBAL_LOAD/STORE path (LOADcnt/STOREcnt only) instead of FLAT (which also
// burns DScnt and the LDS pipe on CDNA5).
typedef const __attribute__((address_space(1))) float* gcf;
typedef       __attribute__((address_space(1))) float* gf;
typedef const __attribute__((address_space(1))) int*   gci;
typedef       __attribute__((address_space(1))) int*   gi;
typedef const __attribute__((address_space(1))) v4f*   gcv4;
typedef       __attribute__((address_space(1))) v4f*   gv4;

__device__ __forceinline__ float lrelu(float y) { return y > 0.0f ? y : 0.2f * y; }

#define WMMA_F32(A, Bf, Cacc) \
  __builtin_amdgcn_wmma_f32_16x16x4_f32(false, (A), false, (Bf), (short)0, (Cacc), false, false)

// ---------------------------------------------------------------- copy (b128)
__global__ void copy_f32x4(float* __restrict__ dst, const float* __restrict__ src, int n4) {
  int i = blockIdx.x * blockDim.x + threadIdx.x;
  if (i < n4) ((gv4)dst)[i] = ((gcv4)src)[i];
}

// ------------------------------------------------ sum of squares over C
// sq[b,s] = sum_c P[b,c,s]^2     P: [B,C,S]
__global__ void sumsq_f32(const float* __restrict__ P, float* __restrict__ sq,
                          int C, int S, int BS) {
  int i = blockIdx.x * blockDim.x + threadIdx.x;
  if (i >= BS) return;
  int b = i / S, s = i - b * S;
  gcf p = (gcf)P + (size_t)b * C * S + s;
  float a = 0.f;
  for (int c = 0; c < C; ++c) { float v = p[(size_t)c * S]; a += v * v; }
  ((gf)sq)[i] = a;
}

// ------------------------------------------------------- GEMM + BN + LReLU
// Y[b,o,n] (+)= lrelu( sum_c W[o,c]*X[b,c, idx? idx[b,n] : n] * sc[o] + sh[o] )
// W:[O,C]  X:[B,C,Nsrc]  idx:[B,N] or null  Y:[B,O,N]
// One wave -> 32(M) x 64(N) tile: two A fragments share four B fragments
// (8 WMMAs per 10 loads per K-step).  8 waves/block cover 512 output columns.
__global__ void gemm_bn_lrelu(const float* __restrict__ Wm, const float* __restrict__ X,
                              const int* __restrict__ idx,
                              const float* __restrict__ sc, const float* __restrict__ sh,
                              float* __restrict__ Y,
                              int O, int C, int N, int Nsrc, int acc) {
  const int lane = threadIdx.x & 31;
  const int wave = threadIdx.x >> 5;
  const int half = lane >> 4;     // 0: K=0,1 / rows 0-7   1: K=2,3 / rows 8-15
  const int l16  = lane & 15;
  const int b  = blockIdx.z;
  const int o0 = blockIdx.y * 32;
  const int n0 = (blockIdx.x * 8 + wave) * 64;
  if (n0 >= N) return;

  gcf Xb = (gcf)X + (size_t)b * C * Nsrc;
  gcf bp[4];                                // running B pointers (K-strided)
#pragma unroll
  for (int j = 0; j < 4; ++j) {
    int n = n0 + 16 * j + l16;
    int col = idx ? ((gci)idx)[b * N + n] : n;
    bp[j] = Xb + (size_t)(2 * half) * Nsrc + col;
  }
  gcf Wr0 = (gcf)Wm + (size_t)(o0 + l16) * C + 2 * half;
  gcf Wr1 = Wr0 + (size_t)16 * C;
  const size_t kstep = 4 * (size_t)Nsrc;

  v8f acc0[4] = {}, acc1[4] = {};
  for (int k = 0; k < C; k += 4) {
    v2f a0; a0.x = Wr0[k]; a0.y = Wr0[k + 1];
    v2f a1; a1.x = Wr1[k]; a1.y = Wr1[k + 1];
#pragma unroll
    for (int j = 0; j < 4; ++j) {
      v2f bf; bf.x = bp[j][0]; bf.y = bp[j][Nsrc];
      acc0[j] = WMMA_F32(a0, bf, acc0[j]);
      acc1[j] = WMMA_F32(a1, bf, acc1[j]);
      bp[j] += kstep;
    }
  }
  gcf scg = (gcf)sc, shg = (gcf)sh;
#pragma unroll
  for (int r = 0; r < 8; ++r) {
    int oA = o0 + r + 8 * half;             // D row for this lane/vgpr (subtile 0)
    int oB = oA + 16;                       // subtile 1
    float sA = scg[oA], tA = shg[oA];
    float sB = scg[oB], tB = shg[oB];
    gf yA = (gf)Y + (size_t)b * O * N + (size_t)oA * N + n0 + l16;
    gf yB = yA + (size_t)16 * N;
#pragma unroll
    for (int j = 0; j < 4; ++j) {
      float y0 = lrelu(acc0[j][r] * sA + tA);
      float y1 = lrelu(acc1[j][r] * sB + tB);
      if (acc) { yA[16 * j] += y0; yB[16 * j] += y1; }
      else     { yA[16 * j]  = y0; yB[16 * j]  = y1; }
    }
  }
}

// ------------------------------------- concat([X1, gather(X2,nn)]) GEMM (always +=)
// Y[b,o,n] += lrelu( sum_{c<2C} W[o,c]*feat(c,n) * sc[o] + sh[o] )
// feat(c,n) = c<C ? X1[b,c,n] : X2[b,c-C, nn[b,n]]
// Two branch-free phases: K in [0,C) reads X1, K in [C,2C) reads gathered X2.
__global__ void gemm_concat_bn_lrelu(const float* __restrict__ Wm,
                                     const float* __restrict__ X1, const float* __restrict__ X2,
                                     const int* __restrict__ nn,
                                     const float* __restrict__ sc, const float* __restrict__ sh,
                                     float* __restrict__ Y,
                                     int O, int C, int N, int S) {
  const int lane = threadIdx.x & 31;
  const int wave = threadIdx.x >> 5;
  const int half = lane >> 4;
  const int l16  = lane & 15;
  const int b  = blockIdx.z;
  const int o0 = blockIdx.y * 32;
  const int n0 = (blockIdx.x * 8 + wave) * 64;
  if (n0 >= N) return;

  gcf Wr0 = (gcf)Wm + (size_t)(o0 + l16) * (2 * C) + 2 * half;
  gcf Wr1 = Wr0 + (size_t)16 * (2 * C);

  v8f acc0[4] = {}, acc1[4] = {};
  gcf bp[4];

  // ---- phase 1: X1 columns (identity)
  gcf X1b = (gcf)X1 + (size_t)b * C * N;
#pragma unroll
  for (int j = 0; j < 4; ++j)
    bp[j] = X1b + (size_t)(2 * half) * N + (n0 + 16 * j + l16);
  {
    const size_t kstep = 4 * (size_t)N;
    for (int k = 0; k < C; k += 4) {
      v2f a0; a0.x = Wr0[k]; a0.y = Wr0[k + 1];
      v2f a1; a1.x = Wr1[k]; a1.y = Wr1[k + 1];
#pragma unroll
      for (int j = 0; j < 4; ++j) {
        v2f bf; bf.x = bp[j][0]; bf.y = bp[j][N];
        acc0[j] = WMMA_F32(a0, bf, acc0[j]);
        acc1[j] = WMMA_F32(a1, bf, acc1[j]);
        bp[j] += kstep;
      }
    }
  }
  // ---- phase 2: X2 columns gathered by nn
  gcf X2b = (gcf)X2 + (size_t)b * C * S;
#pragma unroll
  for (int j = 0; j < 4; ++j)
    bp[j] = X2b + (size_t)(2 * half) * S + ((gci)nn)[b * N + n0 + 16 * j + l16];
  {
    const size_t kstep = 4 * (size_t)S;
    for (int k = 0; k < C; k += 4) {
      v2f a0; a0.x = Wr0[C + k]; a0.y = Wr0[C + k + 1];
      v2f a1; a1.x = Wr1[C + k]; a1.y = Wr1[C + k + 1];
#pragma unroll
      for (int j = 0; j < 4; ++j) {
        v2f bf; bf.x = bp[j][0]; bf.y = bp[j][S];
        acc0[j] = WMMA_F32(a0, bf, acc0[j]);
        acc1[j] = WMMA_F32(a1, bf, acc1[j]);
        bp[j] += kstep;
      }
    }
  }
  gcf scg = (gcf)sc, shg = (gcf)sh;
#pragma unroll
  for (int r = 0; r < 8; ++r) {
    int oA = o0 + r + 8 * half;
    int oB = oA + 16;
    float sA = scg[oA], tA = shg[oA];
    float sB = scg[oB], tB = shg[oB];
    gf yA = (gf)Y + (size_t)b * O * N + (size_t)oA * N + n0 + l16;
    gf yB = yA + (size_t)16 * N;
#pragma unroll
    for (int j = 0; j < 4; ++j) {
      yA[16 * j] += lrelu(acc0[j][r] * sA + tA);
      yB[16 * j] += lrelu(acc1[j][r] * sB + tB);
    }
  }
}

// --------------------------------------------- 1-NN argmin via WMMA distances
// d'[n,s] = ||p2_s||^2 - 2 * <x_n, p2_s>   (||x_n||^2 constant in s -> dropped)
// nn[b,n] = argmin_s d' (first index on ties).  One wave = 16 rows of n,
// two S-tiles per pass sharing one A fragment.
__global__ void nn_argmin(const float* __restrict__ P1,  // [B,C,N]
                          const float* __restrict__ P2,  // [B,C,S]
                          const float* __restrict__ sq,  // [B,S]
                          int* __restrict__ nn,          // [B,N]
                          int C, int N, int S) {
  const int lane = threadIdx.x & 31;
  const int wave = threadIdx.x >> 5;
  const int half = lane >> 4;
  const int l16  = lane & 15;
  const int b  = blockIdx.y;
  const int n0 = (blockIdx.x * 8 + wave) * 16;
  if (n0 >= N) return;

  gcf P1b = (gcf)P1 + (size_t)b * C * N;
  gcf P2b = (gcf)P2 + (size_t)b * C * S;
  gcf sqb = (gcf)sq + (size_t)b * S;
  const size_t kaN = 4 * (size_t)N, kaS = 4 * (size_t)S;

  float best[8]; int bidx[8];
#pragma unroll
  for (int r = 0; r < 8; ++r) { best[r] = 3.4e38f; bidx[r] = 0x7fffffff; }

  for (int s0 = 0; s0 < S; s0 += 32) {
    v8f accA = {}, accB = {};
    gcf pa  = P1b + (size_t)(2 * half) * N + (n0 + l16);
    gcf pb0 = P2b + (size_t)(2 * half) * S + (s0 + l16);
    gcf pb1 = pb0 + 16;
    for (int k = 0; k < C; k += 4) {
      v2f a;   a.x   = pa[0];  a.y   = pa[N];
      v2f bf0; bf0.x = pb0[0]; bf0.y = pb0[S];
      v2f bf1; bf1.x = pb1[0]; bf1.y = pb1[S];
      accA = WMMA_F32(a, bf0, accA);
      accB = WMMA_F32(a, bf1, accB);
      pa += kaN; pb0 += kaS; pb1 += kaS;
    }
    int sA = s0 + l16, sB = sA + 16;         // columns this lane holds
    float svA = sqb[sA], svB = sqb[sB];
#pragma unroll
    for (int r = 0; r < 8; ++r) {
      float dA = svA - 2.0f * accA[r];
      if (dA < best[r] || (dA == best[r] && sA < bidx[r])) { best[r] = dA; bidx[r] = sA; }
      float dB = svB - 2.0f * accB[r];
      if (dB < best[r] || (dB == best[r] && sB < bidx[r])) { best[r] = dB; bidx[r] = sB; }
    }
  }
  // reduce across the 16 lanes of each half (masks 1..8 stay within the half)
#pragma unroll
  for (int off = 1; off < 16; off <<= 1) {
#pragma unroll
    for (int r = 0; r < 8; ++r) {
      float ov = __shfl_xor(best[r], off, 32);
      int   oi = __shfl_xor(bidx[r], off, 32);
      if (ov < best[r] || (ov == best[r] && oi < bidx[r])) { best[r] = ov; bidx[r] = oi; }
    }
  }
  if (l16 == 0) {
#pragma unroll
    for (int r = 0; r < 8; ++r) ((gi)nn)[b * N + n0 + r + 8 * half] = bidx[r];
  }
}

// ================================================================ host side
static void l_copy(hipStream_t st, float* d, const float* s, int n) {
  int n4 = n / 4;
  copy_f32x4<<<(n4 + 255) / 256, 256, 0, st>>>(d, s, n4);
}
static void l_sumsq(hipStream_t st, const float* p, float* sq, int C, int S, int B) {
  int bs = B * S;
  sumsq_f32<<<(bs + 255) / 256, 256, 0, st>>>(p, sq, C, S, bs);
}
static void l_gemm(hipStream_t st, const float* W, const float* X, const int* idx,
                   const float* sc, const float* sh, float* Y,
                   int B, int O, int C, int N, int Nsrc, int acc) {
  dim3 g((N + 511) / 512, O / 32, B);
  gemm_bn_lrelu<<<g, 256, 0, st>>>(W, X, idx, sc, sh, Y, O, C, N, Nsrc, acc);
}
static void l_gemmcat(hipStream_t st, const float* W, const float* X1, const float* X2,
                      const int* nn, const float* sc, const float* sh, float* Y,
                      int B, int O, int C, int N, int S) {
  dim3 g((N + 511) / 512, O / 32, B);
  gemm_concat_bn_lrelu<<<g, 256, 0, st>>>(W, X1, X2, nn, sc, sh, Y, O, C, N, S);
}
static void l_argmin(hipStream_t st, const float* P1, const float* P2, const float* sq,
                     int* nn, int B, int C, int N, int S) {
  dim3 g((N + 127) / 128, B);
  nn_argmin<<<g, 256, 0, st>>>(P1, P2, sq, nn, C, N, S);
}

extern "C" void kernel_launch(void* const* d_in, const int* in_sizes, int n_in,
                              void* d_out, int out_size, void* d_ws, size_t ws_size,
                              hipStream_t stream) {
  (void)in_sizes; (void)n_in; (void)out_size; (void)ws_size;
  const int B = 16;
  const int C1 = 64, C2 = 128, C3 = 256, C4 = 512;
  const int N1 = 4096, N2 = 1024, N3 = 256, N4 = 64;

  const float* b1 = (const float*)d_in[0];
  const float* b2 = (const float*)d_in[1];
  const float* b3 = (const float*)d_in[2];
  const float* b4 = (const float*)d_in[3];
  const int* i2 = (const int*)d_in[4];
  const int* i3 = (const int*)d_in[5];
  const int* i4 = (const int*)d_in[6];

  // params: blocks in insertion order, each {W, s, t}
  enum { S12 = 0, S13, S23, S14, S24, S34, CB1, CB2, CB3, CB4,
         FP43T, FP43M, FP42T, FP42M, FP32T, FP32M,
         FP41T, FP41M, FP31T, FP31M, FP21T, FP21M };
  auto Wp = [&](int i) { return (const float*)d_in[7 + 3 * i]; };
  auto Sp = [&](int i) { return (const float*)d_in[7 + 3 * i + 1]; };
  auto Tp = [&](int i) { return (const float*)d_in[7 + 3 * i + 2]; };

  float* out1 = (float*)d_out;
  float* out2 = out1 + (size_t)B * C1 * N1;
  float* out3 = out2 + (size_t)B * C2 * N2;
  float* out4 = out3 + (size_t)B * C3 * N3;

  float* wsf = (float*)d_ws;
  size_t pos = 0;
  auto alloc = [&](size_t n) { float* p = wsf + pos; pos += n; return p; };

  // ---------------- branch 4: out4 = c4( b4 + s14(g) + s24(g) + s34(g) )
  float* sum4 = alloc((size_t)B * C4 * N4);
  l_copy(stream, sum4, b4, B * C4 * N4);
  l_gemm(stream, Wp(S14), b1, i4, Sp(S14), Tp(S14), sum4, B, C4, C1, N4, N1, 1);
  l_gemm(stream, Wp(S24), b2, i4, Sp(S24), Tp(S24), sum4, B, C4, C2, N4, N2, 1);
  l_gemm(stream, Wp(S34), b3, i4, Sp(S34), Tp(S34), sum4, B, C4, C3, N4, N3, 1);
  l_gemm(stream, Wp(CB4), sum4, nullptr, Sp(CB4), Tp(CB4), out4, B, C4, C4, N4, N4, 0);

  // ---------------- branch 3: out3 = c3( b3 + s13(g) + s23(g) + fp43 )
  float* p43  = alloc((size_t)B * C3 * N4);
  float* sq43 = alloc((size_t)B * N4);
  int*   nn43 = (int*)alloc((size_t)B * N3);
  l_gemm(stream, Wp(FP43T), b4, nullptr, Sp(FP43T), Tp(FP43T), p43, B, C3, C4, N4, N4, 0);
  l_sumsq(stream, p43, sq43, C3, N4, B);
  l_argmin(stream, b3, p43, sq43, nn43, B, C3, N3, N4);
  float* sum3 = alloc((size_t)B * C3 * N3);
  l_copy(stream, sum3, b3, B * C3 * N3);
  l_gemm(stream, Wp(S13), b1, i3, Sp(S13), Tp(S13), sum3, B, C3, C1, N3, N1, 1);
  l_gemm(stream, Wp(S23), b2, i3, Sp(S23), Tp(S23), sum3, B, C3, C2, N3, N2, 1);
  l_gemmcat(stream, Wp(FP43M), b3, p43, nn43, Sp(FP43M), Tp(FP43M), sum3, B, C3, C3, N3, N4);
  l_gemm(stream, Wp(CB3), sum3, nullptr, Sp(CB3), Tp(CB3), out3, B, C3, C3, N3, N3, 0);

  // ---------------- branch 2: out2 = c2( b2 + s12(g) + fp32 + fp42 )
  float* p42  = alloc((size_t)B * C2 * N4);
  float* sq42 = alloc((size_t)B * N4);
  int*   nn42 = (int*)alloc((size_t)B * N2);
  l_gemm(stream, Wp(FP42T), b4, nullptr, Sp(FP42T), Tp(FP42T), p42, B, C2, C4, N4, N4, 0);
  l_sumsq(stream, p42, sq42, C2, N4, B);
  l_argmin(stream, b2, p42, sq42, nn42, B, C2, N2, N4);
  float* p32  = alloc((size_t)B * C2 * N3);
  float* sq32 = alloc((size_t)B * N3);
  int*   nn32 = (int*)alloc((size_t)B * N2);
  l_gemm(stream, Wp(FP32T), b3, nullptr, Sp(FP32T), Tp(FP32T), p32, B, C2, C3, N3, N3, 0);
  l_sumsq(stream, p32, sq32, C2, N3, B);
  l_argmin(stream, b2, p32, sq32, nn32, B, C2, N2, N3);
  float* sum2 = alloc((size_t)B * C2 * N2);
  l_copy(stream, sum2, b2, B * C2 * N2);
  l_gemm(stream, Wp(S12), b1, i2, Sp(S12), Tp(S12), sum2, B, C2, C1, N2, N1, 1);
  l_gemmcat(stream, Wp(FP32M), b2, p32, nn32, Sp(FP32M), Tp(FP32M), sum2, B, C2, C2, N2, N3);
  l_gemmcat(stream, Wp(FP42M), b2, p42, nn42, Sp(FP42M), Tp(FP42M), sum2, B, C2, C2, N2, N4);
  l_gemm(stream, Wp(CB2), sum2, nullptr, Sp(CB2), Tp(CB2), out2, B, C2, C2, N2, N2, 0);

  // ---------------- branch 1: out1 = c1( b1 + fp21 + fp31 + fp41 )
  float* p41  = alloc((size_t)B * C1 * N4);
  float* sq41 = alloc((size_t)B * N4);
  int*   nn41 = (int*)alloc((size_t)B * N1);
  l_gemm(stream, Wp(FP41T), b4, nullptr, Sp(FP41T), Tp(FP41T), p41, B, C1, C4, N4, N4, 0);
  l_sumsq(stream, p41, sq41, C1, N4, B);
  l_argmin(stream, b1, p41, sq41, nn41, B, C1, N1, N4);
  float* p31  = alloc((size_t)B * C1 * N3);
  float* sq31 = alloc((size_t)B * N3);
  int*   nn31 = (int*)alloc((size_t)B * N1);
  l_gemm(stream, Wp(FP31T), b3, nullptr, Sp(FP31T), Tp(FP31T), p31, B, C1, C3, N3, N3, 0);
  l_sumsq(stream, p31, sq31, C1, N3, B);
  l_argmin(stream, b1, p31, sq31, nn31, B, C1, N1, N3);
  float* p21  = alloc((size_t)B * C1 * N2);
  float* sq21 = alloc((size_t)B * N2);
  int*   nn21 = (int*)alloc((size_t)B * N1);
  l_gemm(stream, Wp(FP21T), b2, nullptr, Sp(FP21T), Tp(FP21T), p21, B, C1, C2, N2, N2, 0);
  l_sumsq(stream, p21, sq21, C1, N2, B);
  l_argmin(stream, b1, p21, sq21, nn21, B, C1, N1, N2);
  float* sum1 = alloc((size_t)B * C1 * N1);
  l_copy(stream, sum1, b1, B * C1 * N1);
  l_gemmcat(stream, Wp(FP21M), b1, p21, nn21, Sp(FP21M), Tp(FP21M), sum1, B, C1, C1, N1, N2);
  l_gemmcat(stream, Wp(FP31M), b1, p31, nn31, Sp(FP31M), Tp(FP31M), sum1, B, C1, C1, N1, N3);
  l_gemmcat(stream, Wp(FP41M), b1, p41, nn41, Sp(FP41M), Tp(FP41M), sum1, B, C1, C1, N1, N4);
  l_gemm(stream, Wp(CB1), sum1, nullptr, Sp(CB1), Tp(CB1), out1, B, C1, C1, N1, N1, 0);
}